// Dendrite_755914244692
// MI455X (gfx1250) — compile-verified
//
#include <hip/hip_runtime.h>
#include <math.h>

#define AS1 __attribute__((address_space(1)))
#define AS3 __attribute__((address_space(3)))

typedef int v4i __attribute__((ext_vector_type(4)));

namespace {
constexpr int Bsz = 16, H = 128, W = 128, S = 5;
constexpr int P  = H - S + 1;   // 124 output rows/cols
constexpr int NU = 32;          // dendrite units == lanes of one wave32
constexpr int TI = 8;           // output rows per block (one per wave)
constexpr int TJ = 32;          // output cols per block
constexpr int LR = TI + S - 1;  // 12 LDS tile rows
constexpr int LC = TJ + S - 1;  // 36 LDS tile cols
constexpr float INV_PI = 0.31830988618379067154f;
}

__global__ __launch_bounds__(256)
void dendrite_kernel(const float* __restrict__ x,
                     const float* __restrict__ w,
                     const float* __restrict__ q,
                     float* __restrict__ out) {
  __shared__ __align__(16) float tile[LR * LC];   // 12 x 36 floats = 1728 B

  const int tid  = threadIdx.x;
  const int lane = tid & 31;   // dendrite unit
  const int wave = tid >> 5;   // output row within tile
  const int j0 = blockIdx.x * TJ;
  const int i0 = blockIdx.y * TI;
  const int b  = blockIdx.z;

  // ---- stage x tile into LDS: 108 lanes x 16B via async-to-LDS data mover ----
  if (tid < (LR * LC) / 4) {            // 108 transfers
    const int row = tid / (LC / 4);     // 0..11
    const int seg = tid % (LC / 4);     // 0..8 (16B segments)
    int grow = i0 + row;      if (grow > H - 1) grow = H - 1;   // clamp halo
    int gcol = j0 + seg * 4;  if (gcol > W - 4) gcol = W - 4;   // clamp halo
    const size_t gidx = ((size_t)b * H + grow) * W + gcol;
    const int    lofs = row * LC + seg * 4;
#if __has_builtin(__builtin_amdgcn_global_load_async_to_lds_b128)
    __builtin_amdgcn_global_load_async_to_lds_b128(
        (AS1 v4i*)(x + gidx), (AS3 v4i*)(&tile[lofs]), 0, 0);
#else
    *(float4*)(&tile[lofs]) = *(const float4*)(x + gidx);
#endif
  }
#if __has_builtin(__builtin_amdgcn_global_load_async_to_lds_b128)
  asm volatile("s_wait_asynccnt 0" ::: "memory");
#endif
  __syncthreads();

  // ---- per-lane dendrite weights held in VGPRs ----
  float wv[S * S], qv[S * S];
  {
    const float* wp = w + lane * (S * S);
    const float* qp = q + lane * (S * S);
#pragma unroll
    for (int k = 0; k < S * S; ++k) { wv[k] = wp[k]; qv[k] = qp[k]; }
  }

  const int i = i0 + wave;
  if (i < P) {
#pragma unroll 1
    for (int jj = 0; jj < TJ; ++jj) {
      const int j = j0 + jj;
      if (j < P) {
        float acc = 0.f;
#pragma unroll
        for (int r = 0; r < S; ++r) {
          float prod = 1.f;
#pragma unroll
          for (int c = 0; c < S; ++c) {
            // LDS broadcast read: address uniform across the wave
            const float p = tile[(wave + r) * LC + (jj + c)];
            const float z = __builtin_fmaf(p, wv[r * S + c], -qv[r * S + c]);
            // (pi + 2*atan(10z))/(2pi) + 0.6  ==  1.1 + atan(10z)/pi
            const float t = __builtin_fmaf(atanf(10.f * z), INV_PI, 1.1f);
            prod *= t;
          }
          acc += logf(prod);   // log of row-product, matching reference grouping
        }
        // lane n -> unit n: fully coalesced 128B store per wave
        out[(((size_t)b * P + i) * P + j) * NU + lane] = acc;
      }
    }
  }
}

extern "C" void kernel_launch(void* const* d_in, const int* in_sizes, int n_in,
                              void* d_out, int out_size, void* d_ws, size_t ws_size,
                              hipStream_t stream) {
  (void)in_sizes; (void)n_in; (void)out_size; (void)d_ws; (void)ws_size;
  const float* x = (const float*)d_in[0];
  const float* w = (const float*)d_in[1];
  const float* q = (const float*)d_in[2];
  float* out = (float*)d_out;
  dim3 grid((P + TJ - 1) / TJ,   // 4 column tiles
            (P + TI - 1) / TI,   // 16 row tiles
            Bsz);                // 16 batch images
  dendrite_kernel<<<grid, 256, 0, stream>>>(x, w, q, out);
}